// GINNet_40699110097098
// MI455X (gfx1250) — compile-verified
//
#include <hip/hip_runtime.h>
#include <hip/hip_bf16.h>

#define N_NODES  50000
#define N_EDGES  400000
#define N_GRAPHS 256

typedef float v2f __attribute__((ext_vector_type(2)));
typedef float v8f __attribute__((ext_vector_type(8)));
typedef unsigned u32x4 __attribute__((ext_vector_type(4)));
typedef int      i32x4 __attribute__((ext_vector_type(4)));
typedef int      i32x8 __attribute__((ext_vector_type(8)));

__device__ __forceinline__ void atomic_add_f32(float* p, float v) {
    unsafeAtomicAdd(p, v);
}

// ---------------------------------------------------------------- zero fill
__global__ void k_zero(float* __restrict__ p, long long n) {
    long long i = (long long)blockIdx.x * blockDim.x + threadIdx.x;
    if (i < n) p[i] = 0.0f;
}

// ------------------------------------------------- fused edge MLP + message
// aggr[dst] += relu(x[src] + edge_attr @ Wfc + bfc), per feature.
// One thread handles one edge x 4 features; the 11-wide MLP K-dim lives in
// registers; Wfc columns are L2-resident (<=45KB).
__global__ void k_edge_msg(const float* __restrict__ x,
                           const long long* __restrict__ src,
                           const long long* __restrict__ dst,
                           const float* __restrict__ ea,
                           const float* __restrict__ Wfc,
                           const float* __restrict__ bfc,
                           float* __restrict__ aggr,
                           unsigned total, int D, int nq) {
    unsigned idx = blockIdx.x * blockDim.x + threadIdx.x;
    if (idx >= total) return;
    int e  = (int)(idx / (unsigned)nq);
    int q  = (int)(idx - (unsigned)e * (unsigned)nq);
    int j0 = q << 2;
    long long s = src[e];
    long long d = dst[e];
    float a[11];
#pragma unroll
    for (int k = 0; k < 11; ++k) a[k] = ea[(size_t)e * 11 + k];
#pragma unroll
    for (int jj = 0; jj < 4; ++jj) {
        int j = j0 + jj;
        if (j < D) {
            float v = bfc[j];
#pragma unroll
            for (int k = 0; k < 11; ++k) v = fmaf(a[k], Wfc[k * D + j], v);
            v += x[(size_t)s * D + j];
            v = fmaxf(v, 0.0f);
            atomic_add_f32(&aggr[(size_t)d * D + j], v);
        }
    }
}

// ------------------------------------------------------- fused WMMA f32 GEMM
// Out = elu( ((1+eps)*X + Aggr) @ W + bias ),  X/Aggr: [M,K], W: [K,N].
// Block tile 128(M) x 64(N), 8 waves, each wave computes a 16x64 strip via
// V_WMMA_F32_16X16X4_F32. A is LDS-staged by all waves (float4 path when K
// is a multiple of 16); B is staged by the Tensor Data Mover: wave 0 issues
// TENSOR_LOAD_TO_LDS with a 2-D D# (16 x 64 tile, tensor dims clipped so the
// K tail is zero-filled by OOB reads), waits on TENSORcnt, then the block
// barrier publishes the tile.
#define BM 128
#define BN 64
#define BK 16
#define LDP (BK + 4)   // 80B row stride: 16B-aligned float4 stores, no conflicts

__global__ __launch_bounds__(256) void k_gemm_fused(
        const float* __restrict__ X, const float* __restrict__ Aggr,
        const float* __restrict__ W, const float* __restrict__ bias,
        float* __restrict__ Out, int M, int K, int N,
        const float* __restrict__ epsPtr, int apply_elu) {
    __shared__ float Alds[BM][LDP];    // A[m][k]
    __shared__ float Blds[BK][BN];     // B tile row-major, as TDM writes it

    const float eps_p1 = 1.0f + epsPtr[0];
    const int tid   = threadIdx.x;
    const int wave  = tid >> 5;          // 0..7  -> M sub-row
    const int lane  = tid & 31;
    const int lhalf = lane & 15;
    const int khalf = (lane >> 4) << 1;  // 0 or 2 (K wrap to upper lanes)
    const int m_blk = blockIdx.y * BM;
    const int n_blk = blockIdx.x * BN;
    const int arow  = wave * 16 + lhalf;
    const bool k_vec = ((K & 15) == 0);

    v8f acc[4];
#pragma unroll
    for (int j = 0; j < 4; ++j)
        acc[j] = (v8f){0.f, 0.f, 0.f, 0.f, 0.f, 0.f, 0.f, 0.f};

    for (int k0 = 0; k0 < K; k0 += BK) {
        // ---- B tile: issue async Tensor Data Mover load (wave 0 only) ----
        if (tid < 32) {
            unsigned ldsOff = (unsigned)(size_t)(&Blds[0][0]);
            unsigned long long ga =
                (unsigned long long)(size_t)(W + (size_t)k0 * N + n_blk);
            unsigned td0 = (unsigned)(N - n_blk);   // cols left  (>= 64)
            unsigned td1 = (unsigned)(K - k0);      // rows left  (zero-fill OOB)
            u32x4 g0;
            g0.x = 1u;                               // count=1, user mode
            g0.y = ldsOff;                           // lds_addr (bytes)
            g0.z = (unsigned)ga;                     // global_addr[31:0]
            g0.w = (unsigned)((ga >> 32) & 0x01FFFFFFu) | (2u << 30); // +type=2
            i32x8 g1;
            g1[0] = 0x00020000;                      // data_size = 4 bytes
            g1[1] = (int)(td0 << 16);                // tensor_dim0[15:0]
            g1[2] = (int)((td0 >> 16) | (td1 << 16));// td0 hi | td1 lo
            g1[3] = (int)(((td1 >> 16) & 0xFFFFu) | ((unsigned)BN << 16)); // tile_dim0
            g1[4] = BK;                              // tile_dim1=16, tile_dim2=0
            g1[5] = (int)(unsigned)N;                // tensor_dim0_stride lo
            g1[6] = 0;                               // stride hi
            g1[7] = 0;
            i32x4 gz4 = {0, 0, 0, 0};
            i32x8 gz8 = {0, 0, 0, 0, 0, 0, 0, 0};
            __builtin_amdgcn_tensor_load_to_lds(g0, g1, gz4, gz4, gz8, 0);
        }
        // ---- A tile: fused (1+eps)*X + Aggr, staged by all 8 waves ----
        if (k_vec) {
            for (int i = tid; i < BM * 4; i += 256) {
                int m = i >> 2, kq = (i & 3) << 2;
                int gm = m_blk + m;
                float4 v = make_float4(0.f, 0.f, 0.f, 0.f);
                if (gm < M) {
                    size_t o = (size_t)gm * K + k0 + kq;
                    float4 xv = *(const float4*)(X + o);
                    float4 gv = *(const float4*)(Aggr + o);
                    v.x = fmaf(eps_p1, xv.x, gv.x);
                    v.y = fmaf(eps_p1, xv.y, gv.y);
                    v.z = fmaf(eps_p1, xv.z, gv.z);
                    v.w = fmaf(eps_p1, xv.w, gv.w);
                }
                *(float4*)&Alds[m][kq] = v;
            }
        } else {
            for (int i = tid; i < BM * BK; i += 256) {
                int m = i >> 4, k = i & 15;
                int gm = m_blk + m, gk = k0 + k;
                float v = 0.0f;
                if (gm < M && gk < K) {
                    size_t o = (size_t)gm * K + gk;
                    v = fmaf(eps_p1, X[o], Aggr[o]);
                }
                Alds[m][k] = v;
            }
        }
        if (tid < 32) __builtin_amdgcn_s_wait_tensorcnt(0);
        __syncthreads();

#pragma unroll
        for (int kk = 0; kk < BK; kk += 4) {
            v2f a;
            a.x = Alds[arow][kk + khalf];
            a.y = Alds[arow][kk + khalf + 1];
#pragma unroll
            for (int j = 0; j < 4; ++j) {
                v2f b;
                b.x = Blds[kk + khalf][j * 16 + lhalf];
                b.y = Blds[kk + khalf + 1][j * 16 + lhalf];
                acc[j] = __builtin_amdgcn_wmma_f32_16x16x4_f32(
                    false, a, false, b, (short)0, acc[j], false, false);
            }
        }
        __syncthreads();
    }

    // store: C/D layout -> VGPR r is row (r | r+8), N = lane&15
    const int moff = (lane < 16) ? 0 : 8;
#pragma unroll
    for (int j = 0; j < 4; ++j) {
        int gn = n_blk + j * 16 + lhalf;
        float bv = bias[gn];
#pragma unroll
        for (int r = 0; r < 8; ++r) {
            int gm = m_blk + wave * 16 + moff + r;
            if (gm < M) {
                float v = acc[j][r] + bv;
                if (apply_elu) v = (v > 0.0f) ? v : (expf(v) - 1.0f);
                Out[(size_t)gm * N + gn] = v;
            }
        }
    }
}

// --------------------------------------------------------------- mean pool
__global__ void k_pool_count(const long long* __restrict__ batch,
                             float* __restrict__ counts, int Nn) {
    int n = blockIdx.x * blockDim.x + threadIdx.x;
    if (n < Nn) atomic_add_f32(&counts[batch[n]], 1.0f);
}

__global__ void k_pool_sum(const float* __restrict__ x,
                           const long long* __restrict__ batch,
                           float* __restrict__ out, int Nn) {
    unsigned idx = blockIdx.x * blockDim.x + threadIdx.x;
    if (idx >= (unsigned)Nn * 512u) return;
    int n = (int)(idx >> 9);
    int j = (int)(idx & 511u);
    long long g = batch[n];
    atomic_add_f32(&out[(size_t)g * 512 + j], x[(size_t)n * 512 + j]);
}

__global__ void k_pool_div(float* __restrict__ out,
                           const float* __restrict__ counts, int total) {
    int idx = blockIdx.x * blockDim.x + threadIdx.x;
    if (idx < total) out[idx] /= fmaxf(counts[idx >> 9], 1.0f);
}

// ----------------------------------------------------------------- launcher
static inline unsigned nblk(long long n, int t) { return (unsigned)((n + t - 1) / t); }

extern "C" void kernel_launch(void* const* d_in, const int* in_sizes, int n_in,
                              void* d_out, int out_size, void* d_ws, size_t ws_size,
                              hipStream_t stream) {
    (void)in_sizes; (void)n_in; (void)out_size;
    const float*     x1    = (const float*)d_in[0];
    const long long* ei    = (const long long*)d_in[1];
    const float*     ea    = (const float*)d_in[2];
    const long long* batch = (const long long*)d_in[3];
    const float* W_fc1 = (const float*)d_in[4];  const float* b_fc1 = (const float*)d_in[5];
    const float* W_fc2 = (const float*)d_in[6];  const float* b_fc2 = (const float*)d_in[7];
    const float* W_fc3 = (const float*)d_in[8];  const float* b_fc3 = (const float*)d_in[9];
    const float* W1 = (const float*)d_in[10];    const float* b1 = (const float*)d_in[11];
    const float* W2 = (const float*)d_in[12];    const float* b2 = (const float*)d_in[13];
    const float* W3 = (const float*)d_in[14];    const float* b3 = (const float*)d_in[15];
    const float* eps1 = (const float*)d_in[16];
    const float* eps2 = (const float*)d_in[17];
    const float* eps3 = (const float*)d_in[18];
    float* out = (float*)d_out;

    const long long* src = ei;
    const long long* dst = ei + N_EDGES;

    size_t nA = (size_t)N_NODES * 1024;   // aggr (max width 1024)
    size_t nB = (size_t)N_NODES * 512;    // x ping (512-wide layers)
    size_t nC = (size_t)N_NODES * 1024;   // x pong (1024-wide layer)
    float* bufA   = (float*)d_ws;
    float* bufB   = bufA + nA;
    float* bufC   = bufB + nB;
    float* counts = bufC + nC;
    if (ws_size < (nA + nB + nC + N_GRAPHS) * sizeof(float)) return;

    // ---------------- layer 1: D_in=93 -> 512 ----------------
    {
        long long n = (long long)N_NODES * 93;
        k_zero<<<nblk(n, 256), 256, 0, stream>>>(bufA, n);
        int nq = (93 + 3) >> 2;
        unsigned tot = (unsigned)N_EDGES * (unsigned)nq;
        k_edge_msg<<<nblk(tot, 256), 256, 0, stream>>>(
            x1, src, dst, ea, W_fc1, b_fc1, bufA, tot, 93, nq);
        dim3 g(512 / BN, (N_NODES + BM - 1) / BM);
        k_gemm_fused<<<g, 256, 0, stream>>>(x1, bufA, W1, b1, bufB,
                                            N_NODES, 93, 512, eps1, 1);
    }
    // ---------------- layer 2: 512 -> 1024 -------------------
    {
        long long n = (long long)N_NODES * 512;
        k_zero<<<nblk(n, 256), 256, 0, stream>>>(bufA, n);
        int nq = 512 >> 2;
        unsigned tot = (unsigned)N_EDGES * (unsigned)nq;
        k_edge_msg<<<nblk(tot, 256), 256, 0, stream>>>(
            bufB, src, dst, ea, W_fc2, b_fc2, bufA, tot, 512, nq);
        dim3 g(1024 / BN, (N_NODES + BM - 1) / BM);
        k_gemm_fused<<<g, 256, 0, stream>>>(bufB, bufA, W2, b2, bufC,
                                            N_NODES, 512, 1024, eps2, 1);
    }
    // ---------------- layer 3: 1024 -> 512 -------------------
    {
        long long n = (long long)N_NODES * 1024;
        k_zero<<<nblk(n, 256), 256, 0, stream>>>(bufA, n);
        int nq = 1024 >> 2;
        unsigned tot = (unsigned)N_EDGES * (unsigned)nq;
        k_edge_msg<<<nblk(tot, 256), 256, 0, stream>>>(
            bufC, src, dst, ea, W_fc3, b_fc3, bufA, tot, 1024, nq);
        dim3 g(512 / BN, (N_NODES + BM - 1) / BM);
        k_gemm_fused<<<g, 256, 0, stream>>>(bufC, bufA, W3, b3, bufB,
                                            N_NODES, 1024, 512, eps3, 1);
    }
    // ---------------- global mean pool -----------------------
    {
        long long n = (long long)N_GRAPHS * 512;
        k_zero<<<nblk(n, 256), 256, 0, stream>>>(out, n);
        k_zero<<<1, 256, 0, stream>>>(counts, N_GRAPHS);
        k_pool_count<<<nblk(N_NODES, 256), 256, 0, stream>>>(batch, counts, N_NODES);
        unsigned tot = (unsigned)N_NODES * 512u;
        k_pool_sum<<<nblk(tot, 256), 256, 0, stream>>>(bufB, batch, out, N_NODES);
        k_pool_div<<<nblk(N_GRAPHS * 512, 256), 256, 0, stream>>>(
            out, counts, N_GRAPHS * 512);
    }
}